// GRUEmbedding_61057255080452
// MI455X (gfx1250) — compile-verified
//
#include <hip/hip_runtime.h>
#include <hip/hip_bf16.h>

// Node-subtoken embedding masked-sum (gather + reduce), MI455X / gfx1250.
//
// tokens : [N, 8] int32 subtoken ids     (d_in[0])
// lengths: [N]    int32 valid counts 0..8 (d_in[1])
// emb    : [50000, 256] f32              (d_in[2])  -- 51.2 MB, L2-resident
// out    : [N, 256] f32 = sum_{l<len} emb[tok[l]] * (1/N)
//
// One wave32 per node. Node id made scalar -> token/length fetch via SMEM,
// row gathers are global_load_b128 (saddr uniform + lane*16B), output uses
// non-temporal b128 stores so the 102 MB output stream does not evict the
// L2-resident embedding table.

typedef float v4f __attribute__((ext_vector_type(4)));

#define EMB_DIM 256
#define MAX_SUB 8
#define WAVES_PER_BLOCK 8

__global__ __launch_bounds__(256) void node_emb_sum_kernel(
    const int* __restrict__ tokens,    // [N, 8]
    const int* __restrict__ lengths,   // [N]
    const float* __restrict__ emb,     // [VOCAB, 256]
    float* __restrict__ out,           // [N, 256]
    int n_nodes, float inv_n)
{
    const int lane = (int)(threadIdx.x & 31u);
    int n = (int)(blockIdx.x * WAVES_PER_BLOCK + (threadIdx.x >> 5));
    n = __builtin_amdgcn_readfirstlane(n);   // force SGPR: whole wave = one node
    if (n >= n_nodes) return;

    // Scalar (SMEM) loads: uniform address -> s_load
    int len = lengths[n];
    len = len < 0 ? 0 : (len > MAX_SUB ? MAX_SUB : len);

    int tok[MAX_SUB];
#pragma unroll
    for (int l = 0; l < MAX_SUB; ++l) {
        tok[l] = tokens[n * MAX_SUB + l];
    }

    // Prefetch each valid row into the cache hierarchy. Lane stride of 32 B
    // means one global_prefetch_b8 instruction covers all 8 cachelines of the
    // 1 KiB row (32 lanes x 32 B = 1024 B).
    for (int l = 0; l < len; ++l) {
        __builtin_prefetch(emb + (size_t)tok[l] * EMB_DIM + lane * 8, 0, 3);
    }

    // Gather + accumulate. Each lane owns floats [lane*4 .. lane*4+3] and
    // [128 + lane*4 .. 128 + lane*4+3]: two coalesced global_load_b128 per
    // row, each instruction covering a contiguous 512 B half-row.
    v4f acc0 = {0.f, 0.f, 0.f, 0.f};
    v4f acc1 = {0.f, 0.f, 0.f, 0.f};
    for (int l = 0; l < len; ++l) {
        const v4f* row = (const v4f*)(emb + (size_t)tok[l] * EMB_DIM);
        v4f a = row[lane];
        v4f b = row[lane + 32];
        acc0 += a;
        acc1 += b;
    }

    acc0 *= inv_n;
    acc1 *= inv_n;

    // Non-temporal b128 stores: output is streamed once (102 MB); keep it
    // from evicting the L2-resident embedding table.
    v4f* o = (v4f*)(out + (size_t)n * EMB_DIM);
    __builtin_nontemporal_store(acc0, &o[lane]);
    __builtin_nontemporal_store(acc1, &o[lane + 32]);
}

extern "C" void kernel_launch(void* const* d_in, const int* in_sizes, int n_in,
                              void* d_out, int out_size, void* d_ws, size_t ws_size,
                              hipStream_t stream) {
    const int*   tokens  = (const int*)d_in[0];
    const int*   lengths = (const int*)d_in[1];
    const float* emb     = (const float*)d_in[2];
    float*       out     = (float*)d_out;

    const int n_nodes = in_sizes[1];                 // lengths has N elements
    const float inv_n = 1.0f / (float)n_nodes;       // reference divides by N

    const int blocks = (n_nodes + WAVES_PER_BLOCK - 1) / WAVES_PER_BLOCK;
    node_emb_sum_kernel<<<blocks, 256, 0, stream>>>(tokens, lengths, emb, out,
                                                    n_nodes, inv_n);
}